// Compute_all_A_and_b_matrices_24653112279161
// MI455X (gfx1250) — compile-verified
//
#include <hip/hip_runtime.h>
#include <stdint.h>

// out[e] (4x3) = inv_matrices[e] (4x4) @ conv[mesh_elements[e]] (4x3), E = 2M.
// HBM-bound (~256 MB @ 23.3 TB/s ≈ 11 us floor). CDNA5 paths used:
//  * inv_matrices (128 MB, read-once) streamed by async DMA
//    (global_load_async_to_lds_b128) with double buffering + s_wait_asynccnt.
//  * The 4x4 @ 4x3 math runs on the matrix pipe: one V_WMMA_F32_16X16X4_F32
//    computes 4 elements' products as the diagonal 4x4 blocks of D
//    (A rows m=4e+i carry inv[e][i][k]; B cols n=4e+c carry g[e][k][c]).
//  * NT loads/stores for streaming indices/output keep the 4.8 MB gather
//    table resident in the 192 MB L2.

typedef float v2f __attribute__((ext_vector_type(2)));
typedef float v4f __attribute__((ext_vector_type(4)));
typedef float v8f __attribute__((ext_vector_type(8)));
typedef int   v4i __attribute__((ext_vector_type(4)));

#define TPB      256
#define SLOT_DW  36   // per-thread LDS slot: [0..15] inv, [16..27] g->result, pad to 144 B
#define GRID_MAX 1536

__device__ __forceinline__ void async_copy_b128(uint32_t lds_off, const float* gptr) {
    // GLOBAL_LOAD_ASYNC_TO_LDS_B128, GV mode: vdst = LDS byte offset, vaddr = 64-bit address.
    asm volatile("global_load_async_to_lds_b128 %0, %1, off"
                 :
                 : "v"(lds_off), "v"((uint64_t)(uintptr_t)gptr)
                 : "memory");
}
__device__ __forceinline__ void wait_async_le4() { asm volatile("s_wait_asynccnt 0x4" ::: "memory"); }
__device__ __forceinline__ void wait_async_0()   { asm volatile("s_wait_asynccnt 0x0" ::: "memory"); }
__device__ __forceinline__ void wait_ds_0()      { asm volatile("s_wait_dscnt 0x0"   ::: "memory"); }

__global__ __launch_bounds__(TPB) void fem_elem_kernel(
    const float* __restrict__ conv,    // (N, 3)
    const int*   __restrict__ elems,   // (E, 4)
    const float* __restrict__ invm,    // (E, 4, 4)
    float*       __restrict__ out,     // (E, 4, 3)
    int E)
{
    __shared__ __align__(16) float lds[2][TPB * SLOT_DW];

    const uint32_t tid       = threadIdx.x;
    const uint32_t lane      = tid & 31u;       // wave32
    const uint32_t waveSlot0 = tid & ~31u;      // first slot of this wave's 32 elements

    const long long tileStride = (long long)gridDim.x * TPB;
    const long long base0      = (long long)blockIdx.x * TPB;
    if (base0 >= E) return;

    auto issue = [&](long long base, int buf) {
        const long long e = base + tid;
        if (e < (long long)E) {
            const float* g = invm + e * 16;
            const uint32_t l = (uint32_t)(uintptr_t)(&lds[buf][tid * SLOT_DW]);
            async_copy_b128(l + 0u,  g + 0);
            async_copy_b128(l + 16u, g + 4);
            async_copy_b128(l + 32u, g + 8);
            async_copy_b128(l + 48u, g + 12);
        }
    };

    // Lane roles for the diagonal-block WMMA (constant per thread):
    const uint32_t n     = lane & 15u;          // matrix column / A row (mod 16)
    const uint32_t e_loc = n >> 2;              // which of the 4 elements in a WMMA group
    const uint32_t ci    = n & 3u;              // column within element (3 = unused)
    const uint32_t irow  = lane & 3u;           // A-row within element
    const uint32_t k0    = (lane < 16u) ? 0u : 2u;  // K pair held by this lane half
    const bool     hi    = (lane & 4u) != 0u;       // result rows in D[4..7] vs D[0..3]
    const bool     useful = ((lane < 8u) || (lane >= 24u)) && (ci != 3u);

    int buf = 0;
    issue(base0, 0);   // prime the async pipeline

    for (long long base = base0; base < (long long)E; base += tileStride) {
        const bool more = (base + tileStride) < (long long)E;
        if (more) issue(base + tileStride, buf ^ 1);

        const long long e = base + tid;
        const bool active = (e < (long long)E);

        // Indices: streaming, read-once -> non-temporal b128 load.
        v4i q = {0, 0, 0, 0};
        if (active) q = __builtin_nontemporal_load((const v4i*)(elems + e * 4));

        // Gather 4 vertex vectors (regular cached loads; table lives in L2).
        float g0x = 0.f, g0y = 0.f, g0z = 0.f;
        float g1x = 0.f, g1y = 0.f, g1z = 0.f;
        float g2x = 0.f, g2y = 0.f, g2z = 0.f;
        float g3x = 0.f, g3y = 0.f, g3z = 0.f;
        if (active) {
            const float* p0 = conv + 3ll * q.x;
            const float* p1 = conv + 3ll * q.y;
            const float* p2 = conv + 3ll * q.z;
            const float* p3 = conv + 3ll * q.w;
            g0x = p0[0]; g0y = p0[1]; g0z = p0[2];
            g1x = p1[0]; g1y = p1[1]; g1z = p1[2];
            g2x = p2[0]; g2y = p2[1]; g2z = p2[2];
            g3x = p3[0]; g3y = p3[1]; g3z = p3[2];
        }

        float* slotMe = &lds[buf][tid * SLOT_DW];
        if (active) {
            // Stage gathered vectors transposed [c][j] so the WMMA B operand
            // (K-pair k0,k0+1 of column c) is one aligned ds_load_b64.
            *(v4f*)(slotMe + 16) = (v4f){g0x, g1x, g2x, g3x};   // c = 0
            *(v4f*)(slotMe + 20) = (v4f){g0y, g1y, g2y, g3y};   // c = 1
            *(v4f*)(slotMe + 24) = (v4f){g0z, g1z, g2z, g3z};   // c = 2
        }

        // Drain this tile's async DMA (inv now in LDS); also a compiler barrier
        // between the cross-lane g-stores above and the DS reads below.
        if (more) wait_async_le4();
        else      wait_async_0();

        const bool fullWave = ((long long)(base + waveSlot0) + 32ll) <= (long long)E;

        if (fullWave) {
            // ---- matrix-pipe path: 8 WMMAs cover this wave's 32 elements ----
            // Per lane: A = inv[e][irow][k0..k0+1], B = g[e][k0..k0+1][ci],
            // where e is the e_loc-th element of WMMA group g.
            const float* grpBase = &lds[buf][(waveSlot0 + e_loc) * SLOT_DW];
#pragma unroll
            for (int gi = 0; gi < 8; ++gi) {
                const float* sa = grpBase + (size_t)gi * 4 * SLOT_DW;
                const v2f A = *(const v2f*)(sa + irow * 4 + k0);       // 8B aligned
                const v2f B = *(const v2f*)(sa + 16 + ci * 4 + k0);    // 8B aligned
                const v8f C = {0.f, 0.f, 0.f, 0.f, 0.f, 0.f, 0.f, 0.f};
                // D = A(16x4) x B(4x16) + 0 ; diagonal 4x4 blocks = 4 elements' results
                v8f D = __builtin_amdgcn_wmma_f32_16x16x4_f32(
                            false, A, false, B, (short)0, C, false, false);
                if (useful) {
                    // This lane owns column ci of element e_loc: rows are
                    // D[0..3] or D[4..7] depending on (lane & 4).
                    v4f col;
                    col.x = hi ? D[4] : D[0];
                    col.y = hi ? D[5] : D[1];
                    col.z = hi ? D[6] : D[2];
                    col.w = hi ? D[7] : D[3];
                    *(v4f*)((float*)sa + 16 + ci * 4) = col;   // result in [c][i] layout
                }
            }

            wait_ds_0();   // cross-lane result scatter -> own-slot readback

            const v4f rc0 = *(const v4f*)(slotMe + 16);   // out[i][0], i=0..3
            const v4f rc1 = *(const v4f*)(slotMe + 20);   // out[i][1]
            const v4f rc2 = *(const v4f*)(slotMe + 24);   // out[i][2]

            float* op = out + e * 12;
            const v4f o0 = {rc0[0], rc1[0], rc2[0], rc0[1]};
            const v4f o1 = {rc1[1], rc2[1], rc0[2], rc1[2]};
            const v4f o2 = {rc2[2], rc0[3], rc1[3], rc2[3]};
            __builtin_nontemporal_store(o0, (v4f*)(op + 0));
            __builtin_nontemporal_store(o1, (v4f*)(op + 4));
            __builtin_nontemporal_store(o2, (v4f*)(op + 8));
        } else if (active) {
            // ---- VALU tail path (partial wave: WMMA needs EXEC all-ones) ----
            const v4f a0 = *(const v4f*)(slotMe + 0);
            const v4f a1 = *(const v4f*)(slotMe + 4);
            const v4f a2 = *(const v4f*)(slotMe + 8);
            const v4f a3 = *(const v4f*)(slotMe + 12);

            const float r0x = a0.x*g0x + a0.y*g1x + a0.z*g2x + a0.w*g3x;
            const float r0y = a0.x*g0y + a0.y*g1y + a0.z*g2y + a0.w*g3y;
            const float r0z = a0.x*g0z + a0.y*g1z + a0.z*g2z + a0.w*g3z;
            const float r1x = a1.x*g0x + a1.y*g1x + a1.z*g2x + a1.w*g3x;
            const float r1y = a1.x*g0y + a1.y*g1y + a1.z*g2y + a1.w*g3y;
            const float r1z = a1.x*g0z + a1.y*g1z + a1.z*g2z + a1.w*g3z;
            const float r2x = a2.x*g0x + a2.y*g1x + a2.z*g2x + a2.w*g3x;
            const float r2y = a2.x*g0y + a2.y*g1y + a2.z*g2y + a2.w*g3y;
            const float r2z = a2.x*g0z + a2.y*g1z + a2.z*g2z + a2.w*g3z;
            const float r3x = a3.x*g0x + a3.y*g1x + a3.z*g2x + a3.w*g3x;
            const float r3y = a3.x*g0y + a3.y*g1y + a3.z*g2y + a3.w*g3y;
            const float r3z = a3.x*g0z + a3.y*g1z + a3.z*g2z + a3.w*g3z;

            float* op = out + e * 12;
            const v4f o0 = {r0x, r0y, r0z, r1x};
            const v4f o1 = {r1y, r1z, r2x, r2y};
            const v4f o2 = {r2z, r3x, r3y, r3z};
            __builtin_nontemporal_store(o0, (v4f*)(op + 0));
            __builtin_nontemporal_store(o1, (v4f*)(op + 4));
            __builtin_nontemporal_store(o2, (v4f*)(op + 8));
        }

        buf ^= 1;
    }
}

extern "C" void kernel_launch(void* const* d_in, const int* in_sizes, int n_in,
                              void* d_out, int out_size, void* d_ws, size_t ws_size,
                              hipStream_t stream) {
    const float* conv  = (const float*)d_in[0];   // (N, 3)   f32
    const int*   elems = (const int*)  d_in[1];   // (E, 4)   i32
    const float* invm  = (const float*)d_in[2];   // (E, 4, 4) f32
    float*       out   = (float*)d_out;           // (E, 4, 3) f32

    const int E = in_sizes[1] / 4;
    int tiles = (E + TPB - 1) / TPB;
    int grid  = tiles < GRID_MAX ? tiles : GRID_MAX;
    if (grid < 1) grid = 1;

    fem_elem_kernel<<<grid, TPB, 0, stream>>>(conv, elems, invm, out, E);
}